// LabelWiseMLC_26534307955166
// MI455X (gfx1250) — compile-verified
//
#include <hip/hip_runtime.h>

typedef __attribute__((ext_vector_type(2))) float v2f;
typedef __attribute__((ext_vector_type(8))) float v8f;

#define B_DIM 128
#define L_DIM 512
#define I_DIM 1024

// One wave computes out[b0..b0+15, l] via V_WMMA_F32_16X16X4_F32.
// A tile: 16 rows (batches) x 4 K-chunk of doc_rep (per-lane float2 load).
// B tile: W[l, kchunk..kchunk+3] broadcast across all 16 N columns
//         (per-lane float2 load of a lane-half-uniform address).
// D: all 16 columns identical == the 16 dot products for this label.
__global__ __launch_bounds__(256) void labelwise_mlc_wmma(
    const float* __restrict__ X,     // (B, L, I)
    const float* __restrict__ W,     // (L, I)
    const float* __restrict__ bias,  // (L,)
    float* __restrict__ out)         // (B, L)
{
  const int lane = threadIdx.x & 31;
  const int wave = threadIdx.x >> 5;
  const int tile = blockIdx.x * 8 + wave;   // 0..4095
  const int l    = tile & (L_DIM - 1);      // label index
  const int b0   = (tile >> 9) << 4;        // batch tile base (tile/512 * 16)

  const int m  = lane & 15;                 // A-matrix row within tile
  const int hi = lane >> 4;                 // lane half -> K sub-offset 2*hi

  const float* arow = X + ((size_t)(b0 + m) * L_DIM + l) * I_DIM + 2 * hi;
  const float* wrow = W + (size_t)l * I_DIM + 2 * hi;

  v8f acc0 = {}, acc1 = {}, acc2 = {}, acc3 = {};

  for (int k = 0; k < I_DIM; k += 16) {
    v2f a0 = *(const v2f*)(arow + k);
    v2f w0 = *(const v2f*)(wrow + k);
    v2f a1 = *(const v2f*)(arow + k + 4);
    v2f w1 = *(const v2f*)(wrow + k + 4);
    v2f a2 = *(const v2f*)(arow + k + 8);
    v2f w2 = *(const v2f*)(wrow + k + 8);
    v2f a3 = *(const v2f*)(arow + k + 12);
    v2f w3 = *(const v2f*)(wrow + k + 12);

    acc0 = __builtin_amdgcn_wmma_f32_16x16x4_f32(false, a0, false, w0,
                                                 (short)0, acc0, false, false);
    acc1 = __builtin_amdgcn_wmma_f32_16x16x4_f32(false, a1, false, w1,
                                                 (short)0, acc1, false, false);
    acc2 = __builtin_amdgcn_wmma_f32_16x16x4_f32(false, a2, false, w2,
                                                 (short)0, acc2, false, false);
    acc3 = __builtin_amdgcn_wmma_f32_16x16x4_f32(false, a3, false, w3,
                                                 (short)0, acc3, false, false);
  }

  v8f acc = (acc0 + acc1) + (acc2 + acc3);
  const float bl = bias[l];

  // D layout (fp32 16x16 C/D): vgpr v of lane holds M = v + 8*hi, N = lane%16.
  // All N columns equal -> lanes 0 and 16 cover all 16 batch rows.
  if (m == 0) {
#pragma unroll
    for (int v = 0; v < 8; ++v) {
      float z = acc[v] + bl;
      float s = 1.0f / (1.0f + expf(-z));
      out[(size_t)(b0 + 8 * hi + v) * L_DIM + l] = s;
    }
  }
}

extern "C" void kernel_launch(void* const* d_in, const int* in_sizes, int n_in,
                              void* d_out, int out_size, void* d_ws, size_t ws_size,
                              hipStream_t stream) {
  const float* X    = (const float*)d_in[0];  // doc_rep (128, 512, 1024) f32
  const float* W    = (const float*)d_in[1];  // W       (512, 1024)      f32
  const float* bias = (const float*)d_in[2];  // b       (512,)           f32
  float* out        = (float*)d_out;          // (128, 512) f32

  (void)in_sizes; (void)n_in; (void)out_size; (void)d_ws; (void)ws_size;

  // 4096 waves total: 512 blocks x 8 waves (256 threads, wave32).
  labelwise_mlc_wmma<<<512, 256, 0, stream>>>(X, W, bias, out);
}